// DarcyFlowOperator_35407710388663
// MI455X (gfx1250) — compile-verified
//
#include <hip/hip_runtime.h>

// ---------------------------------------------------------------------------
// Darcy-flow graph PDE residual for MI455X (gfx1250, wave32).
// Bandwidth-bound sparse scatter: uses b128 NT streaming loads,
// global_prefetch_b8 (CDNA5 prefetch path) and global_atomic_add_f32
// (atomics resolve in the 192MB L2; node arrays are only 4MB each).
// ---------------------------------------------------------------------------

typedef __attribute__((ext_vector_type(4))) int   v4i_t;
typedef __attribute__((ext_vector_type(4))) float v4f_t;

// ---- zero the accumulator region of the workspace (vec4 stores + tail) ----
__global__ __launch_bounds__(256)
void darcy_zero(float* __restrict__ p, long n) {
  const long tid = (long)blockIdx.x * blockDim.x + threadIdx.x;
  const long nth = (long)gridDim.x * blockDim.x;
  const long n4  = n >> 2;
  v4f_t z = (v4f_t)0.0f;
  for (long i = tid; i < n4; i += nth)
    __builtin_nontemporal_store(z, ((v4f_t*)p) + i);
  for (long i = (n4 << 2) + tid; i < n; i += nth)
    p[i] = 0.0f;
}

// ---- one edge sweep: fused x-derivative and y-derivative accumulation -----
// gX/gY: gather sources for x-edges / y-edges (pass1: both = x,
// pass2: tmp0 / tmp1). do_count=1 only on pass 1 (counts are identical
// between passes, so pass 2 reuses them -> two fewer atomic streams).
__global__ __launch_bounds__(256)
void darcy_edge_pass(const int*   __restrict__ ei,    // [2*E], src then dst
                     const float* __restrict__ attr,  // [E,2] interleaved
                     const float* __restrict__ gX,
                     const float* __restrict__ gY,
                     float* __restrict__ sumX, float* __restrict__ cntX,
                     float* __restrict__ sumY, float* __restrict__ cntY,
                     int E, int do_count) {
  const long nvec = (long)(E >> 2);
  const long tid  = (long)blockIdx.x * blockDim.x + threadIdx.x;
  const long nth  = (long)gridDim.x * blockDim.x;
  const v4i_t* sI = (const v4i_t*)ei;          // 4 src indices per elem
  const v4i_t* dI = (const v4i_t*)(ei + E);    // 4 dst indices per elem
  const v4f_t* aV = (const v4f_t*)attr;        // 2 elems cover 4 edges

  for (long v = tid; v < nvec; v += nth) {
    // CDNA5 prefetch of next grid-stride iteration's cache lines
    if (v + nth < nvec) {
      __builtin_prefetch((const void*)(sI + v + nth), 0, 0);
      __builtin_prefetch((const void*)(dI + v + nth), 0, 0);
      __builtin_prefetch((const void*)(aV + 2 * (v + nth)), 0, 0);
    }
    // non-temporal b128 streaming loads (edge data is touched once/pass)
    v4i_t s4  = __builtin_nontemporal_load(sI + v);
    v4i_t d4  = __builtin_nontemporal_load(dI + v);
    v4f_t a01 = __builtin_nontemporal_load(aV + 2 * v);      // e0.xy e1.xy
    v4f_t a23 = __builtin_nontemporal_load(aV + 2 * v + 1);  // e2.xy e3.xy

    const float ax[4] = {a01.x, a01.z, a23.x, a23.z};
    const float ay[4] = {a01.y, a01.w, a23.y, a23.w};

#pragma unroll
    for (int j = 0; j < 4; ++j) {
      const int si = s4[j];
      const int di = d4[j];
      const float a0 = ax[j];
      const float a1 = ay[j];
      if (a0 != 0.0f) {                       // x-edge
        const float m = (gX[si] - gX[di]) / a0;
        atomicAdd(&sumX[di], m);
        if (do_count) atomicAdd(&cntX[di], 1.0f);
      }
      if (a1 != 0.0f) {                       // y-edge
        const float m = (gY[si] - gY[di]) / a1;
        atomicAdd(&sumY[di], m);
        if (do_count) atomicAdd(&cntY[di], 1.0f);
      }
    }
  }

  // scalar tail (E not a multiple of 4)
  for (long e = (nvec << 2) + tid; e < (long)E; e += nth) {
    const int si = ei[e];
    const int di = ei[(long)E + e];
    const float a0 = attr[2 * e];
    const float a1 = attr[2 * e + 1];
    if (a0 != 0.0f) {
      const float m = (gX[si] - gX[di]) / a0;
      atomicAdd(&sumX[di], m);
      if (do_count) atomicAdd(&cntX[di], 1.0f);
    }
    if (a1 != 0.0f) {
      const float m = (gY[si] - gY[di]) / a1;
      atomicAdd(&sumY[di], m);
      if (do_count) atomicAdd(&cntY[di], 1.0f);
    }
  }
}

// ---- mid node pass: tmp_flow = a_x * mean-derivative --------------------
__global__ __launch_bounds__(256)
void darcy_node_mid(const float* __restrict__ a_x,
                    const float* __restrict__ sumX, const float* __restrict__ cntX,
                    const float* __restrict__ sumY, const float* __restrict__ cntY,
                    float* __restrict__ tmp0, float* __restrict__ tmp1, int N) {
  const int i = blockIdx.x * blockDim.x + threadIdx.x;
  if (i < N) {
    const float a = a_x[i];
    tmp0[i] = a * (sumX[i] / fmaxf(cntX[i], 1.0f));
    tmp1[i] = a * (sumY[i] / fmaxf(cntY[i], 1.0f));
  }
}

// ---- final node pass: divergence + forcing, masked ----------------------
__global__ __launch_bounds__(256)
void darcy_node_final(const float* __restrict__ sum2X, const float* __restrict__ cntX,
                      const float* __restrict__ sum2Y, const float* __restrict__ cntY,
                      const int* __restrict__ mask, float* __restrict__ out, int N) {
  const int i = blockIdx.x * blockDim.x + threadIdx.x;
  if (i < N) {
    const float dxx = sum2X[i] / fmaxf(cntX[i], 1.0f);
    const float dyy = sum2Y[i] / fmaxf(cntY[i], 1.0f);
    out[i] = (dxx + dyy + 1.0f) * (1.0f - (float)mask[i]);
  }
}

extern "C" void kernel_launch(void* const* d_in, const int* in_sizes, int n_in,
                              void* d_out, int out_size, void* d_ws, size_t ws_size,
                              hipStream_t stream) {
  const float* x    = (const float*)d_in[0];   // [N,1] f32
  const float* a_x  = (const float*)d_in[1];   // [N,1] f32
  const int*   ei   = (const int*)  d_in[2];   // [2,E] i32
  const float* ea   = (const float*)d_in[3];   // [E,2] f32
  const int*   mask = (const int*)  d_in[4];   // [N]   i32
  float* out = (float*)d_out;

  const int N = in_sizes[0];
  const int E = in_sizes[2] / 2;

  // workspace layout: 8 * N floats = 32 MB at N = 1M
  float* ws    = (float*)d_ws;
  float* sumX  = ws + 0L * N;
  float* cntX  = ws + 1L * N;
  float* sumY  = ws + 2L * N;
  float* cntY  = ws + 3L * N;
  float* sum2X = ws + 4L * N;
  float* sum2Y = ws + 5L * N;
  float* tmp0  = ws + 6L * N;
  float* tmp1  = ws + 7L * N;

  const int T = 256;
  const int nodeBlocks = (N + T - 1) / T;
  // grid-stride edge kernels: enough waves to fill the chip, a few
  // iterations each so the prefetch distance is meaningful.
  const long nvec = (long)(E >> 2);
  int edgeBlocks = (int)((nvec + T - 1) / T);
  if (edgeBlocks > 4096) edgeBlocks = 4096;
  if (edgeBlocks < 1) edgeBlocks = 1;

  // 1) zero the 6N accumulator floats (sums + counts); tmp0/1 fully written.
  {
    long nzero = 6L * N;
    int zb = (int)((nzero / 4 + T - 1) / T);
    if (zb > 4096) zb = 4096;
    if (zb < 1) zb = 1;
    darcy_zero<<<zb, T, 0, stream>>>(ws, nzero);
  }

  // 2) first derivative sweep (dx, dy) with counts
  darcy_edge_pass<<<edgeBlocks, T, 0, stream>>>(
      ei, ea, x, x, sumX, cntX, sumY, cntY, E, /*do_count=*/1);

  // 3) tmp_flow = a_x * nabla2d
  darcy_node_mid<<<nodeBlocks, T, 0, stream>>>(
      a_x, sumX, cntX, sumY, cntY, tmp0, tmp1, N);

  // 4) divergence sweep (dxx from tmp0 on x-edges, dyy from tmp1 on y-edges),
  //    counts reused from pass 1
  darcy_edge_pass<<<edgeBlocks, T, 0, stream>>>(
      ei, ea, tmp0, tmp1, sum2X, sumX /*unused cnt sink? no: */, sum2Y, sumY,
      E, /*do_count=*/0);

  // 5) pde_loss = (dxx + dyy + f) * (1 - mask)
  darcy_node_final<<<nodeBlocks, T, 0, stream>>>(
      sum2X, cntX, sum2Y, cntY, mask, out, N);
}